// LocallyDirected1D_61263413510614
// MI455X (gfx1250) — compile-verified
//
#include <hip/hip_runtime.h>

#define N_IN    100000
#define N_OUT   5000
#define N_EDGES 200000
#define FILTERS 4
#define BATCH   128

typedef __attribute__((ext_vector_type(2))) float v2f;
typedef __attribute__((ext_vector_type(8))) float v8f;

// ---------------- zero the histogram bins (ws is not re-poisoned between calls)
__global__ void ld1d_zero_counts(int* __restrict__ counts) {
    int i = blockIdx.x * blockDim.x + threadIdx.x;
    if (i < N_OUT) counts[i] = 0;
}

// ---------------- x (BATCH x N_IN) -> xT (N_IN x BATCH), LDS-tiled so both sides coalesce
__global__ void ld1d_transpose(const float* __restrict__ x, float* __restrict__ xT) {
    __shared__ float tile[32][33];
    int iBase = blockIdx.x * 32;   // N_IN dimension
    int bBase = blockIdx.y * 32;   // batch dimension (128 % 32 == 0)
    int tx = threadIdx.x, ty = threadIdx.y; // block (32, 8)
#pragma unroll
    for (int r = 0; r < 4; ++r) {
        int row = ty + r * 8;
        int i = iBase + tx;
        tile[row][tx] = (i < N_IN) ? x[(size_t)(bBase + row) * N_IN + i] : 0.0f;
    }
    __syncthreads();
#pragma unroll
    for (int r = 0; r < 4; ++r) {
        int row = ty + r * 8;
        int i = iBase + row;
        if (i < N_IN) xT[(size_t)i * BATCH + (bBase + tx)] = tile[tx][row];
    }
}

// ---------------- counting sort of edges by output column: histogram
__global__ void ld1d_hist(const int* __restrict__ cols, int* __restrict__ counts) {
    int e = blockIdx.x * blockDim.x + threadIdx.x;
    if (e < N_EDGES) atomicAdd(&counts[cols[e]], 1);
}

// ---------------- exclusive scan over 5000 bins (one workgroup)
__global__ void ld1d_scan(const int* __restrict__ counts, int* __restrict__ segStart,
                          int* __restrict__ cursor) {
    __shared__ int partial[256];
    const int CH = 20; // 256*20 = 5120 >= 5000
    int t = threadIdx.x;
    int base = t * CH;
    int local[CH];
    int s = 0;
#pragma unroll
    for (int i = 0; i < CH; ++i) {
        int idx = base + i;
        int v = (idx < N_OUT) ? counts[idx] : 0;
        local[i] = s;
        s += v;
    }
    partial[t] = s;
    __syncthreads();
    if (t == 0) {
        int acc = 0;
        for (int j = 0; j < 256; ++j) { int v = partial[j]; partial[j] = acc; acc += v; }
        segStart[N_OUT] = acc; // == N_EDGES
    }
    __syncthreads();
    int off = partial[t];
#pragma unroll
    for (int i = 0; i < CH; ++i) {
        int idx = base + i;
        if (idx < N_OUT) { int p = off + local[i]; segStart[idx] = p; cursor[idx] = p; }
    }
}

// ---------------- place edge ids into their column segment
__global__ void ld1d_scatter(const int* __restrict__ cols, int* __restrict__ cursor,
                             int* __restrict__ sorted) {
    int e = blockIdx.x * blockDim.x + threadIdx.x;
    if (e < N_EDGES) {
        int pos = atomicAdd(&cursor[cols[e]], 1);
        sorted[pos] = e;
    }
}

// ---------------- per-column segmented GEMM via V_WMMA_F32_16X16X4_F32
// One workgroup per output column; 8 wave32s, wave w owns batches [16w, 16w+16) (M).
// A (16x4 f32, 2 VGPR): lane<16 -> K={k,k+1}, lane>=16 -> K={k+2,k+3}, M = lane&15.
// B (4x16 f32, 2 VGPR): N = lane&15 (filters in N=0..3, rest zero-padded), same K split.
// C/D (16x16 f32, 8 VGPR): M = vgpr + 8*(lane>=16), N = lane&15.
// Chunks are padded to a multiple of 8 with {offset=0, weights=0} entries so the
// K-loop is completely branch/predication free (EXEC all-1s at every WMMA).
// Gathers use a uniform SGPR base + 32-bit VGPR element index so the backend can
// emit global_load_b32 vD, vIdx, s[base] scale_offset (no per-lane 64-bit adds).
#define CHUNK 128
__global__ void __launch_bounds__(256)
ld1d_compute(const float* __restrict__ xT, const float* __restrict__ weight,
             const float* __restrict__ bias, const int* __restrict__ rows,
             const int* __restrict__ segStart, const int* __restrict__ sorted,
             float* __restrict__ out) {
    __shared__ __align__(16) int   s_off[CHUNK];       // row*BATCH element offsets (padded: 0)
    __shared__ __align__(16) float s_w16[CHUNK * 16];  // B rows padded to N=16; cols 4..15 stay 0

    int c    = blockIdx.x;
    int tid  = threadIdx.x;
    int lane = tid & 31;
    int wave = tid >> 5;
    int m    = lane & 15;              // A: M index / B,D: N index
    int half = lane >> 4;              // K-pair selector
    unsigned lanebase = (unsigned)(wave * 16 + m);  // per-lane column of xT

    // zero entire padded-B region once; cols 4..15 are never written again
    for (int i = tid; i < CHUNK * 16; i += 256) s_w16[i] = 0.0f;

    int beg = segStart[c];
    int end = segStart[c + 1];

    v8f acc = {0.f, 0.f, 0.f, 0.f, 0.f, 0.f, 0.f, 0.f};

    for (int basee = beg; basee < end; basee += CHUNK) {
        int cnt = end - basee;
        if (cnt > CHUNK) cnt = CHUNK;
        int cntPad = (cnt + 7) & ~7;

        __syncthreads();
        for (int i = tid; i < cntPad; i += 256) {
            int    roff = 0;
            float4 w4   = {0.f, 0.f, 0.f, 0.f};
            if (i < cnt) {
                int e = sorted[basee + i];
                roff = rows[e] * BATCH;
                w4   = ((const float4*)weight)[e];
            }
            s_off[i] = roff;
            *((float4*)(s_w16 + i * 16)) = w4;   // ds_store_b128
        }
        if (tid == 0 && basee + CHUNK < end)
            __builtin_prefetch(&sorted[basee + CHUNK], 0, 1); // global_prefetch_b8
        __syncthreads();

        for (int k = 0; k < cntPad; k += 8) {
            int q0 = (k >> 1) + half;           // K pair {2q0, 2q0+1}
            int q1 = q0 + 2;                    // K pair for the k+4 step
            int2 o0 = ((const int2*)s_off)[q0]; // ds_load_b64
            int2 o1 = ((const int2*)s_off)[q1];

            v2f b0, b1, a0, a1;
            b0.x = s_w16[(q0 * 2    ) * 16 + m];
            b0.y = s_w16[(q0 * 2 + 1) * 16 + m];
            b1.x = s_w16[(q1 * 2    ) * 16 + m];
            b1.y = s_w16[(q1 * 2 + 1) * 16 + m];

            // uniform base + 32-bit index -> saddr + voffset (scale_offset) form
            a0.x = xT[(unsigned)o0.x + lanebase];
            a0.y = xT[(unsigned)o0.y + lanebase];
            a1.x = xT[(unsigned)o1.x + lanebase];
            a1.y = xT[(unsigned)o1.y + lanebase];

            // 8 args: (neg_a, A, neg_b, B, c_mod, C, reuse_a, reuse_b)
            acc = __builtin_amdgcn_wmma_f32_16x16x4_f32(
                false, a0, false, b0, (short)0, acc, false, false);
            acc = __builtin_amdgcn_wmma_f32_16x16x4_f32(
                false, a1, false, b1, (short)0, acc, false, false);
        }
    }

    // epilogue: bias + ReLU; lanes with N<4 own real filters
    if (m < FILTERS) {
        float bv = bias[c * FILTERS + m];
        int mb = wave * 16 + half * 8;
#pragma unroll
        for (int j = 0; j < 8; ++j) {
            float v = acc[j] + bv;
            v = v > 0.f ? v : 0.f;
            out[(size_t)(mb + j) * (N_OUT * FILTERS) + c * FILTERS + m] = v;
        }
    }
}

static inline size_t ld1d_align256(size_t v) { return (v + 255) & ~(size_t)255; }

extern "C" void kernel_launch(void* const* d_in, const int* in_sizes, int n_in,
                              void* d_out, int out_size, void* d_ws, size_t ws_size,
                              hipStream_t stream) {
    const float* x      = (const float*)d_in[0]; // (128, 100000)
    const float* weight = (const float*)d_in[1]; // (200000, 4)
    const float* bias   = (const float*)d_in[2]; // (5000, 4)
    const int*   rows   = (const int*)d_in[3];   // (200000,)
    const int*   cols   = (const int*)d_in[4];   // (200000,)
    float* out = (float*)d_out;                  // (128, 5000, 4)

    // workspace carve-up (~52.1 MB)
    char* ws = (char*)d_ws;
    float* xT      = (float*)ws; ws += ld1d_align256((size_t)N_IN * BATCH * sizeof(float));
    int*   counts  = (int*)ws;   ws += ld1d_align256((size_t)N_OUT * sizeof(int));
    int*   segSt   = (int*)ws;   ws += ld1d_align256((size_t)(N_OUT + 1) * sizeof(int));
    int*   cursor  = (int*)ws;   ws += ld1d_align256((size_t)N_OUT * sizeof(int));
    int*   sorted  = (int*)ws;   ws += ld1d_align256((size_t)N_EDGES * sizeof(int));

    ld1d_zero_counts<<<(N_OUT + 255) / 256, 256, 0, stream>>>(counts);
    ld1d_transpose<<<dim3((N_IN + 31) / 32, BATCH / 32), dim3(32, 8), 0, stream>>>(x, xT);
    ld1d_hist<<<(N_EDGES + 255) / 256, 256, 0, stream>>>(cols, counts);
    ld1d_scan<<<1, 256, 0, stream>>>(counts, segSt, cursor);
    ld1d_scatter<<<(N_EDGES + 255) / 256, 256, 0, stream>>>(cols, cursor, sorted);
    ld1d_compute<<<N_OUT, 256, 0, stream>>>(xT, weight, bias, rows, segSt, sorted, out);
}